// ContrastiveLossL2_10033043604035
// MI455X (gfx1250) — compile-verified
//
#include <hip/hip_runtime.h>
#include <math.h>

typedef __attribute__((ext_vector_type(2))) float v2f;
typedef __attribute__((ext_vector_type(8))) float v8f;

#define B_ 8
#define N_ 307200
#define D_ 16
#define M_ 5000
#define K_ 50000

// workspace float-index layout
#define WS_MATCH 0   // 1 float: sum of squared match diffs
#define WS_DSUM  1   // 8 floats: per-batch sum of non-match dists
#define WS_NMSUM 9   // 8 floats: per-batch sum of hinge^2
#define WS_HN    17  // 8 floats: per-batch hard-negative count
#define WS_ACC_COUNT 32
#define WS_DIST  32  // B*K floats: staged distances (optional)

__global__ __launch_bounds__(32) void zero_acc_kernel(float* ws) {
    ws[threadIdx.x] = 0.0f;  // zero ws[0..31]
}

// ---------------------------------------------------------------------------
// Match loss: one wave handles 16 pairs (2 lanes per pair, 8 dims per lane).
// Sum of all squared diffs via V_WMMA_F32_16X16X4_F32 with A = diff^2,
// B = ones: every D element is a row-sum of A, so sum(D) = 16 * sum(diff^2).
// A total sum is layout-invariant, so lane->matrix-slot mapping is free.
// 40000 pairs / 16 per wave = 2500 waves = 625 blocks of 128.
// ---------------------------------------------------------------------------
__global__ __launch_bounds__(128) void match_wmma_kernel(
    const float* __restrict__ outA, const float* __restrict__ outB,
    const int* __restrict__ matchA, const int* __restrict__ matchB,
    float* __restrict__ ws)
{
    const int tid  = threadIdx.x;
    const int lane = tid & 31;
    const int wv   = tid >> 5;
    const int gw   = blockIdx.x * 4 + wv;      // global wave id, [0, 2500)
    const int q    = gw * 16 + (lane >> 1);    // global pair id, [0, 40000)
    const int half = lane & 1;                 // which 8 dims this lane owns

    const int b  = q / M_;
    const int ia = matchA[q];
    const int ib = matchB[q];

    const float4* pa = (const float4*)(outA + ((size_t)b * N_ + (size_t)ia) * D_ + 8 * half);
    const float4* pb = (const float4*)(outB + ((size_t)b * N_ + (size_t)ib) * D_ + 8 * half);
    float4 a0 = pa[0], a1 = pa[1];
    float4 b0 = pb[0], b1 = pb[1];

    float d2[8];
    d2[0] = (a0.x - b0.x) * (a0.x - b0.x);
    d2[1] = (a0.y - b0.y) * (a0.y - b0.y);
    d2[2] = (a0.z - b0.z) * (a0.z - b0.z);
    d2[3] = (a0.w - b0.w) * (a0.w - b0.w);
    d2[4] = (a1.x - b1.x) * (a1.x - b1.x);
    d2[5] = (a1.y - b1.y) * (a1.y - b1.y);
    d2[6] = (a1.z - b1.z) * (a1.z - b1.z);
    d2[7] = (a1.w - b1.w) * (a1.w - b1.w);

    v2f ones; ones.x = 1.0f; ones.y = 1.0f;
    v8f c = {};
#pragma unroll
    for (int ch = 0; ch < 4; ++ch) {
        v2f a; a.x = d2[2 * ch]; a.y = d2[2 * ch + 1];
        // D = A x ones + C : chained f32 WMMA accumulation (8 args form)
        c = __builtin_amdgcn_wmma_f32_16x16x4_f32(
                false, a, false, ones, (short)0, c, false, false);
    }

    float s = c[0] + c[1] + c[2] + c[3] + c[4] + c[5] + c[6] + c[7];
#pragma unroll
    for (int off = 16; off > 0; off >>= 1) s += __shfl_down(s, off, 32);

    __shared__ float ls[4];
    if (lane == 0) ls[wv] = s;
    __syncthreads();
    if (tid == 0) {
        float tile = (ls[0] + ls[1] + ls[2] + ls[3]) * (1.0f / 16.0f);
        atomicAdd(&ws[WS_MATCH], tile);
    }
}

// ---------------------------------------------------------------------------
// Block reduction helper: 256 threads (8 waves), result valid on thread 0.
// ---------------------------------------------------------------------------
__device__ __forceinline__ float block_reduce_256(float v, float* ls) {
    const int lane = threadIdx.x & 31;
    const int wv   = threadIdx.x >> 5;
#pragma unroll
    for (int off = 16; off > 0; off >>= 1) v += __shfl_down(v, off, 32);
    if (lane == 0) ls[wv] = v;
    __syncthreads();
    float r = 0.0f;
    if (wv == 0) {
        r = (lane < 8) ? ls[lane] : 0.0f;
#pragma unroll
        for (int off = 4; off > 0; off >>= 1) r += __shfl_down(r, off, 32);
    }
    return r;
}

__device__ __forceinline__ float gather_dist(
    const float* __restrict__ outA, const float* __restrict__ outB,
    int b, int ia, int ib)
{
    const float4* pa = (const float4*)(outA + ((size_t)b * N_ + (size_t)ia) * D_);
    const float4* pb = (const float4*)(outB + ((size_t)b * N_ + (size_t)ib) * D_);
    float s = 0.0f;
#pragma unroll
    for (int j = 0; j < 4; ++j) {
        float4 a = pa[j], bb = pb[j];
        float dx = a.x - bb.x, dy = a.y - bb.y, dz = a.z - bb.z, dw = a.w - bb.w;
        s += dx * dx + dy * dy + dz * dz + dw * dw;
    }
    return sqrtf(s);
}

// Pass 1: distances + per-batch sum (for mean). Optionally stage dist in ws.
template <bool STORE>
__global__ __launch_bounds__(256) void nm_dist_kernel(
    const float* __restrict__ outA, const float* __restrict__ outB,
    const int* __restrict__ nmA, const int* __restrict__ nmB,
    float* __restrict__ ws)
{
    const int b = blockIdx.y;
    const int k = blockIdx.x * 256 + threadIdx.x;
    float dist = 0.0f;
    if (k < K_) {
        const int ia = nmA[b * K_ + k];
        const int ib = nmB[b * K_ + k];
        dist = gather_dist(outA, outB, b, ia, ib);
        if (STORE) ws[WS_DIST + (size_t)b * K_ + k] = dist;
    }
    __shared__ float ls[8];
    float bs = block_reduce_256(dist, ls);
    if (threadIdx.x == 0) atomicAdd(&ws[WS_DSUM + b], bs);
}

// Pass 2: hinge loss vs the (now known) per-batch mean + hard-negative count.
template <bool RECOMPUTE>
__global__ __launch_bounds__(256) void nm_loss_kernel(
    const float* __restrict__ outA, const float* __restrict__ outB,
    const int* __restrict__ nmA, const int* __restrict__ nmB,
    float* __restrict__ ws)
{
    const int b = blockIdx.y;
    const int k = blockIdx.x * 256 + threadIdx.x;
    const float mean = ws[WS_DSUM + b] * (1.0f / (float)K_);

    float nm = 0.0f, cnt = 0.0f;
    if (k < K_) {
        float d;
        if (RECOMPUTE) {
            d = gather_dist(outA, outB, b, nmA[b * K_ + k], nmB[b * K_ + k]);
        } else {
            d = ws[WS_DIST + (size_t)b * K_ + k];
        }
        float t = mean - d;
        nm  = (t > 0.0f) ? t * t : 0.0f;
        cnt = (nm > 0.0f) ? 1.0f : 0.0f;
    }
    __shared__ float ls0[8];
    __shared__ float ls1[8];
    float snm = block_reduce_256(nm, ls0);
    __syncthreads();
    float scn = block_reduce_256(cnt, ls1);
    if (threadIdx.x == 0) {
        atomicAdd(&ws[WS_NMSUM + b], snm);
        atomicAdd(&ws[WS_HN + b], scn);
    }
}

__global__ __launch_bounds__(32) void finalize_kernel(
    const float* __restrict__ ws, const int* __restrict__ hardNeg,
    float* __restrict__ out)
{
    const int lane = threadIdx.x;
    float nml = 0.0f;
    if (lane < B_) {
        const float hn   = ws[WS_HN + lane];
        const int   hard = hardNeg[0];
        const float denom = (hard != 0) ? ((hn == 0.0f) ? (float)K_ : hn)
                                        : (float)K_;
        nml = ws[WS_NMSUM + lane] / denom;  // NON_MATCH_LOSS_WEIGHT == 1.0
    }
#pragma unroll
    for (int off = 16; off > 0; off >>= 1) nml += __shfl_down(nml, off, 32);
    if (lane == 0) {
        const float match_sum = ws[WS_MATCH] * (1.0f / (float)M_);
        out[0] = match_sum + nml;
        out[1] = match_sum;
        out[2] = nml;
    }
}

extern "C" void kernel_launch(void* const* d_in, const int* in_sizes, int n_in,
                              void* d_out, int out_size, void* d_ws, size_t ws_size,
                              hipStream_t stream) {
    (void)in_sizes; (void)n_in; (void)out_size;
    const float* outA   = (const float*)d_in[0];
    const float* outB   = (const float*)d_in[1];
    const int*   matchA = (const int*)d_in[2];
    const int*   matchB = (const int*)d_in[3];
    const int*   nmA    = (const int*)d_in[4];
    const int*   nmB    = (const int*)d_in[5];
    const int*   hard   = (const int*)d_in[6];
    float* out = (float*)d_out;
    float* ws  = (float*)d_ws;

    const size_t need = (size_t)(WS_ACC_COUNT + (size_t)B_ * K_) * sizeof(float);
    const bool stage = (ws_size >= need);  // stage dist[B,K] (1.6MB, L2-resident)

    zero_acc_kernel<<<1, 32, 0, stream>>>(ws);

    match_wmma_kernel<<<(B_ * M_) / (16 * 4), 128, 0, stream>>>(
        outA, outB, matchA, matchB, ws);

    dim3 g((K_ + 255) / 256, B_);
    if (stage) {
        nm_dist_kernel<true ><<<g, 256, 0, stream>>>(outA, outB, nmA, nmB, ws);
        nm_loss_kernel<false><<<g, 256, 0, stream>>>(outA, outB, nmA, nmB, ws);
    } else {
        nm_dist_kernel<false><<<g, 256, 0, stream>>>(outA, outB, nmA, nmB, ws);
        nm_loss_kernel<true ><<<g, 256, 0, stream>>>(outA, outB, nmA, nmB, ws);
    }

    finalize_kernel<<<1, 32, 0, stream>>>(ws, hard, out);
}